// CombinedLoss_6949257085514
// MI455X (gfx1250) — compile-verified
//
#include <hip/hip_runtime.h>
#include <hip/hip_bf16.h>

typedef __attribute__((ext_vector_type(2))) float v2f;
typedef __attribute__((ext_vector_type(8))) float v8f;

#define NPTS 4096
#define NB 4
#define ROWS_PER_BLOCK 128      // 8 waves * 16 rows
#define BM_HI (NPTS + 16)       // 32-bank pad between the two B planes
#define BM_ELEMS (2 * NPTS + 16)
#define SMEM_BYTES (BM_ELEMS * sizeof(v2f) + NPTS * 3 * sizeof(float))

// Symmetric per-round min/argmin kernel (ROUND is compile-time).
// Rows from X, columns from Y: per row r, argmin_c [ |Y_c|^2 - 2<X_r,Y_c> ]
// (row-constant |X_r|^2 dropped: argmin-invariant; restored for round-0 sums).
// masked(r,c) = (c == selfMask[k][b][r]) || (r == otherMask[k][b][c]), k < ROUND.
template <int ROUND>
__global__ __launch_bounds__(256) void min_round_kernel(
    const float* __restrict__ X, const float* __restrict__ Y,
    const unsigned short* __restrict__ selfMask,   // [4][B][NPTS]
    const unsigned short* __restrict__ otherMask,  // [4][B][NPTS]
    unsigned short* __restrict__ outIdx,           // [B][NPTS] for this round
    float* accum)
{
    extern __shared__ char smem[];
    v2f*   bm  = (v2f*)smem;                                  // [2 planes][NPTS]
    float* raw = (float*)(smem + BM_ELEMS * sizeof(v2f));     // [NPTS*3]

    const int blocksPerBatch = NPTS / ROWS_PER_BLOCK;
    const int b = blockIdx.x / blocksPerBatch;
    const int rowBlock = (blockIdx.x % blocksPerBatch) * ROWS_PER_BLOCK;
    const size_t BN = (size_t)NB * NPTS;

    const float* Xb = X + (size_t)b * NPTS * 3;
    const float* Yb = Y + (size_t)b * NPTS * 3;

    // ---- Stage raw column points (48 KB) via gfx1250 async global->LDS DMA.
    {
        const unsigned rawOff = (unsigned)(uintptr_t)raw;     // LDS byte offset
        for (int i = threadIdx.x; i < NPTS * 3 / 4; i += 256) {
            const float* g = Yb + i * 4;
            unsigned ldsOff = rawOff + i * 16;
            asm volatile("global_load_async_to_lds_b128 %0, %1, off"
                         :: "v"(ldsOff), "v"(g) : "memory");
        }
        asm volatile("s_wait_asynccnt 0x0" ::: "memory");
    }
    __syncthreads();

    // ---- Transform to WMMA B-operand layout:
    // plane 0 (lanes 0-15):  K0,K1 = (-2x, -2y); plane 1 (lanes 16-31): K2,K3 = (-2z, |y|^2)
    for (int i = threadIdx.x; i < NPTS; i += 256) {
        float yx = raw[3 * i + 0];
        float yy = raw[3 * i + 1];
        float yz = raw[3 * i + 2];
        v2f lo, hi;
        lo.x = -2.0f * yx; lo.y = -2.0f * yy;
        hi.x = -2.0f * yz; hi.y = yx * yx + yy * yy + yz * yz;
        bm[i] = lo;
        bm[BM_HI + i] = hi;
    }
    __syncthreads();

    const int wave = threadIdx.x >> 5;     // wave32
    const int lane = threadIdx.x & 31;
    const int laneHalf = lane >> 4;
    const int lcol = lane & 15;
    const int rowBase = rowBlock + wave * 16;

    // A-matrix (16x4 f32): lane L holds row L%16; VGPR0=K(0|2), VGPR1=K(1|3).
    const int arow = rowBase + lcol;
    float ax = Xb[3 * arow + 0];
    float ay = Xb[3 * arow + 1];
    float az = Xb[3 * arow + 2];
    v2f A;
    A.x = laneHalf ? az : ax;
    A.y = laneHalf ? 1.0f : ay;

    // C/D layout: lane L, VGPR v <-> row (v + 8*laneHalf), col L%16.
    int myrow[8];
    int smv[8][ROUND ? ROUND : 1];
    #pragma unroll
    for (int v = 0; v < 8; ++v) {
        myrow[v] = rowBase + v + laneHalf * 8;
        if constexpr (ROUND > 0) {
            #pragma unroll
            for (int k = 0; k < ROUND; ++k)
                smv[v][k] = selfMask[(size_t)k * BN + (size_t)b * NPTS + myrow[v]];
        }
    }

    const unsigned short* omB = otherMask + (size_t)b * NPTS;
    const v2f* bmHalf = bm + laneHalf * BM_HI;

    float minv[8];
    int   mini[8];
    #pragma unroll
    for (int v = 0; v < 8; ++v) { minv[v] = __builtin_inff(); mini[v] = 0; }

    #pragma unroll 2
    for (int cb = 0; cb < NPTS; cb += 16) {
        const int c = cb + lcol;
        v2f Bm = bmHalf[c];

        v8f acc = {};
        acc = __builtin_amdgcn_wmma_f32_16x16x4_f32(
                  false, A, false, Bm, (short)0, acc, false, false);

        int om[ROUND ? ROUND : 1];
        if constexpr (ROUND > 0) {
            #pragma unroll
            for (int k = 0; k < ROUND; ++k)
                om[k] = omB[(size_t)k * BN + c];
        }

        #pragma unroll
        for (int v = 0; v < 8; ++v) {
            float d = acc[v];
            if constexpr (ROUND > 0) {
                bool masked = false;
                #pragma unroll
                for (int k = 0; k < ROUND; ++k)
                    masked = masked || (c == smv[v][k]) || (myrow[v] == om[k]);
                if (masked) d = __builtin_inff();
            }
            bool lt = d < minv[v];
            mini[v] = lt ? c : mini[v];
            minv[v] = lt ? d : minv[v];
        }
    }

    // argmin reduction across the 16 lanes of each half.
    #pragma unroll
    for (int off = 1; off < 16; off <<= 1) {
        #pragma unroll
        for (int v = 0; v < 8; ++v) {
            float ov = __shfl_xor(minv[v], off, 32);
            int   oi = __shfl_xor(mini[v], off, 32);
            if (ov < minv[v] || (ov == minv[v] && oi < mini[v])) {
                minv[v] = ov; mini[v] = oi;
            }
        }
    }

    if (lcol == 0) {
        float rsum = 0.0f;
        #pragma unroll
        for (int v = 0; v < 8; ++v) {
            const int r = myrow[v];
            outIdx[(size_t)b * NPTS + r] = (unsigned short)mini[v];
            if constexpr (ROUND == 0) {
                float xx = Xb[3 * r + 0];
                float xy = Xb[3 * r + 1];
                float xz = Xb[3 * r + 2];
                rsum += minv[v] + xx * xx + xy * xy + xz * xz;
            }
        }
        if constexpr (ROUND == 0) atomicAdd(accum, rsum);
    }
    (void)accum;
}

// Normal-loss partial sums: S[(k-1)*B + b] = sum_m <p0 - p_k, normal[b][m]>
__global__ __launch_bounds__(256) void normal_partial_kernel(
    const float* __restrict__ preds, const float* __restrict__ normals,
    const unsigned short* __restrict__ i_f2s,   // [4][B][NPTS]
    float* __restrict__ S)
{
    const int gid = blockIdx.x * 256 + threadIdx.x;
    const int b = gid / NPTS;        // uniform per block
    const int m = gid % NPTS;
    const size_t BN = (size_t)NB * NPTS;
    const float* Pb = preds + (size_t)b * NPTS * 3;

    const int i0 = i_f2s[0 * BN + (size_t)b * NPTS + m];
    float p0x = Pb[3 * i0 + 0], p0y = Pb[3 * i0 + 1], p0z = Pb[3 * i0 + 2];
    float nx = normals[(size_t)b * NPTS * 3 + 3 * m + 0];
    float ny = normals[(size_t)b * NPTS * 3 + 3 * m + 1];
    float nz = normals[(size_t)b * NPTS * 3 + 3 * m + 2];

    float s[3];
    #pragma unroll
    for (int k = 1; k <= 3; ++k) {
        const int ik = i_f2s[(size_t)k * BN + (size_t)b * NPTS + m];
        float ex = p0x - Pb[3 * ik + 0];
        float ey = p0y - Pb[3 * ik + 1];
        float ez = p0z - Pb[3 * ik + 2];
        s[k - 1] = ex * nx + ey * ny + ez * nz;
    }

    #pragma unroll
    for (int off = 16; off >= 1; off >>= 1)
        #pragma unroll
        for (int j = 0; j < 3; ++j)
            s[j] += __shfl_xor(s[j], off, 32);

    if ((threadIdx.x & 31) == 0)
        #pragma unroll
        for (int j = 0; j < 3; ++j)
            atomicAdd(&S[j * NB + b], s[j]);
}

__global__ void init_accum_kernel(float* acc) {
    if (threadIdx.x < 13) acc[threadIdx.x] = 0.0f;   // accum + S[12]
}

__global__ void finalize_kernel(const float* accum, const float* S, float* out) {
    float nl = 0.0f;
    for (int i = 0; i < 12; ++i) nl += fabsf(S[i]);
    out[0] = 1.0f + accum[0] + 10.0f * nl;
}

extern "C" void kernel_launch(void* const* d_in, const int* in_sizes, int n_in,
                              void* d_out, int out_size, void* d_ws, size_t ws_size,
                              hipStream_t stream) {
    const float* gts     = (const float*)d_in[0];
    const float* preds   = (const float*)d_in[1];
    const float* normals = (const float*)d_in[2];

    char* ws = (char*)d_ws;
    const size_t planeBytes = (size_t)NB * NPTS * sizeof(unsigned short); // 32 KB
    unsigned short* i_f2s = (unsigned short*)ws;                          // [4][B][N]
    unsigned short* i_s2f = (unsigned short*)(ws + 4 * planeBytes);       // [4][B][N]
    float* accum = (float*)(ws + 8 * planeBytes);                         // 1 float
    float* S     = accum + 1;                                             // 12 floats

    init_accum_kernel<<<1, 32, 0, stream>>>(accum);

    const int grid = NB * (NPTS / ROWS_PER_BLOCK);   // 128 blocks
    const size_t shmem = SMEM_BYTES;                 // ~112 KB (<=320 KB/WGP)

    #define LAUNCH_ROUND(R)                                                        \
        do {                                                                       \
            min_round_kernel<R><<<grid, 256, shmem, stream>>>(                     \
                gts, preds, i_s2f, i_f2s, i_s2f + (size_t)R * NB * NPTS, accum);   \
            min_round_kernel<R><<<grid, 256, shmem, stream>>>(                     \
                preds, gts, i_f2s, i_s2f, i_f2s + (size_t)R * NB * NPTS, accum);   \
        } while (0)

    LAUNCH_ROUND(0);
    LAUNCH_ROUND(1);
    LAUNCH_ROUND(2);
    LAUNCH_ROUND(3);
    #undef LAUNCH_ROUND

    normal_partial_kernel<<<NB * NPTS / 256, 256, 0, stream>>>(
        preds, normals, i_f2s, S);

    finalize_kernel<<<1, 1, 0, stream>>>(accum, S, (float*)d_out);
}